// QuantumQuanvolutionClassifier_65481071398800
// MI455X (gfx1250) — compile-verified
//
#include <hip/hip_runtime.h>
#include <hip/hip_bf16.h>
#include <math.h>

// ---------------------------------------------------------------------------
// 4-qubit statevector helpers. Wire w <-> bit (3-w) of the 16-dim index
// (wire 0 = MSB), matching the reference's reshape/moveaxis convention.
// Everything is templated/unrolled so the 16-float state lives in VGPRs.
// ---------------------------------------------------------------------------

template <int W>
__device__ __forceinline__ void apply_ry(float* s, float c, float sn) {
  constexpr int bit = 1 << (3 - W);
#pragma unroll
  for (int i0 = 0; i0 < 16; ++i0) {
    if (i0 & bit) continue;
    const int i1 = i0 | bit;
    const float a = s[i0], b = s[i1];
    s[i0] = c * a - sn * b;
    s[i1] = sn * a + c * b;
  }
}

template <int C, int T>
__device__ __forceinline__ void apply_cx(float* s) {
  constexpr int cbit = 1 << (3 - C);
  constexpr int tbit = 1 << (3 - T);
  float t[16];
#pragma unroll
  for (int i = 0; i < 16; ++i) t[i] = s[i];
#pragma unroll
  for (int i = 0; i < 16; ++i) s[i] = t[(i & cbit) ? (i ^ tbit) : i];
}

// encode = |0000> + RY(a_w) on each wire: a product state, computed directly.
__device__ __forceinline__ void encode_state(float* s, const float* ang) {
  const float c0 = __cosf(0.5f * ang[0]), s0 = __sinf(0.5f * ang[0]);
  const float c1 = __cosf(0.5f * ang[1]), s1 = __sinf(0.5f * ang[1]);
  const float c2 = __cosf(0.5f * ang[2]), s2 = __sinf(0.5f * ang[2]);
  const float c3 = __cosf(0.5f * ang[3]), s3 = __sinf(0.5f * ang[3]);
#pragma unroll
  for (int i = 0; i < 16; ++i) {
    float v = ((i >> 3) & 1) ? s0 : c0;   // wire 0 -> bit 3
    v *= ((i >> 2) & 1) ? s1 : c1;        // wire 1 -> bit 2
    v *= ((i >> 1) & 1) ? s2 : c2;        // wire 2 -> bit 1
    v *= (i & 1) ? s3 : c3;               // wire 3 -> bit 0
    s[i] = v;
  }
}

// ---------------------------------------------------------------------------
// Kernel 1: one thread per 2x2 patch. Quanv circuit -> feats[4], then the
// attention circuit (feats re-encoded as angles) -> raw scalar.
// ---------------------------------------------------------------------------
__global__ void __launch_bounds__(256)
quanv_kernel(const float* __restrict__ x, const float* __restrict__ rl,
             float* __restrict__ feats, float* __restrict__ raw, int total) {
  const int p = blockIdx.x * blockDim.x + threadIdx.x;
  if (p >= total) return;
  const int b = p / 196;
  const int pp = p - b * 196;
  const int py = pp / 14, px = pp - py * 14;
  const float* img = x + (size_t)b * 784;
  const int r0 = 2 * py, c0 = 2 * px;
  float ang[4];
  ang[0] = img[r0 * 28 + c0];
  ang[1] = img[r0 * 28 + c0 + 1];
  ang[2] = img[(r0 + 1) * 28 + c0];
  ang[3] = img[(r0 + 1) * 28 + c0 + 1];

  float rc[5], rs[5];
#pragma unroll
  for (int i = 0; i < 5; ++i) {
    rc[i] = __cosf(0.5f * rl[i]);
    rs[i] = __sinf(0.5f * rl[i]);
  }

  float s[16];
  encode_state(s, ang);
  apply_ry<0>(s, rc[0], rs[0]);
  apply_ry<1>(s, rc[1], rs[1]);
  apply_cx<0, 2>(s);
  apply_ry<2>(s, rc[2], rs[2]);
  apply_ry<3>(s, rc[3], rs[3]);
  apply_cx<1, 3>(s);
  apply_ry<0>(s, rc[4], rs[4]);
  apply_cx<2, 3>(s);

  float f[4] = {0.f, 0.f, 0.f, 0.f};
#pragma unroll
  for (int i = 0; i < 16; ++i) {
    const float q = s[i] * s[i];
    f[0] += ((i >> 3) & 1) ? -q : q;
    f[1] += ((i >> 2) & 1) ? -q : q;
    f[2] += ((i >> 1) & 1) ? -q : q;
    f[3] += (i & 1) ? -q : q;
  }
  *(float4*)(feats + (size_t)p * 4) = make_float4(f[0], f[1], f[2], f[3]);

  // attention circuit: encode(feats), CX01, CX12, CX23, <Z0>
  float s2[16];
  encode_state(s2, f);
  apply_cx<0, 1>(s2);
  apply_cx<1, 2>(s2);
  apply_cx<2, 3>(s2);
  float z0 = 0.f;
#pragma unroll
  for (int i = 0; i < 16; ++i) {
    const float q = s2[i] * s2[i];
    z0 += ((i >> 3) & 1) ? -q : q;
  }
  raw[p] = z0;
}

// ---------------------------------------------------------------------------
// Kernel 2: one block per image. Softmax over 196 raws; scale feats in place
// -> weighted (B, 784) with the reference (patch, feature) flattening.
// ---------------------------------------------------------------------------
__global__ void __launch_bounds__(256)
attn_kernel(const float* __restrict__ raw, float* __restrict__ feats) {
  __shared__ float red[256];
  const int b = blockIdx.x;
  const int t = threadIdx.x;
  const float v = (t < 196) ? raw[b * 196 + t] : -INFINITY;
  red[t] = v;
  __syncthreads();
#pragma unroll
  for (int off = 128; off > 0; off >>= 1) {
    if (t < off) red[t] = fmaxf(red[t], red[t + off]);
    __syncthreads();
  }
  const float m = red[0];
  __syncthreads();
  const float e = (t < 196) ? __expf(v - m) : 0.f;
  red[t] = e;
  __syncthreads();
#pragma unroll
  for (int off = 128; off > 0; off >>= 1) {
    if (t < off) red[t] += red[t + off];
    __syncthreads();
  }
  const float inv = 1.0f / red[0];
  if (t < 196) {
    const float a = e * inv;
    float4* fp = (float4*)(feats + ((size_t)b * 196 + t) * 4);
    float4 f = *fp;
    f.x *= a; f.y *= a; f.z *= a; f.w *= a;
    *fp = f;
  }
}

// ---------------------------------------------------------------------------
// Kernel 3: WMMA f32 GEMM. logits(Bn,16) = weighted(Bn,784) @ Wpad^T(784,16).
// One wave computes a 16x16 tile via V_WMMA_F32_16X16X4_F32, K in steps of 4.
// W (10x784) is staged into LDS padded to 16 rows (zeros) with row stride 788
// floats (16 lanes hit 16 distinct banks). 4 waves/block, 4 tiles/block.
//
// Fragment layouts per ISA 7.12.2:
//   A 16x4 f32 : lane L -> M=L%16, VGPR{0,1} = K = (L<16 ? {0,1} : {2,3}) + k
//   B 4x16 f32 : lane L -> N=L%16, VGPR{0,1} = K = (L<16 ? {0,1} : {2,3}) + k
//   C/D 16x16  : VGPR r -> M = r + 8*(L>=16), N = L%16
// ---------------------------------------------------------------------------
typedef __attribute__((ext_vector_type(2))) float v2f;
typedef __attribute__((ext_vector_type(8))) float v8f;

#define WSTRIDE 788

__global__ void __launch_bounds__(128)
gemm_wmma_kernel(const float* __restrict__ A, const float* __restrict__ W,
                 float* __restrict__ logits) {
  __shared__ float lw[16 * WSTRIDE];
  const int tid = threadIdx.x;
  for (int i = tid; i < 16 * WSTRIDE; i += 128) lw[i] = 0.f;
  __syncthreads();
  for (int i = tid; i < 10 * 784; i += 128) {
    const int n = i / 784;
    const int k = i - n * 784;
    lw[n * WSTRIDE + k] = W[i];
  }
  __syncthreads();

  const int wave = tid >> 5;
  const int lane = tid & 31;
  const int tile = blockIdx.x * 4 + wave;
  const int mrow = (tile << 4) + (lane & 15);
  const int khalf = (lane >> 4) << 1;  // 0 for lanes 0-15, 2 for lanes 16-31
  const float* arow = A + (size_t)mrow * 784 + khalf;
  const float* brow = lw + (lane & 15) * WSTRIDE + khalf;

  v8f acc = {};
#pragma unroll 4
  for (int k = 0; k < 784; k += 4) {
    const v2f a = *(const v2f*)(arow + k);
    const v2f bm = *(const v2f*)(brow + k);
    acc = __builtin_amdgcn_wmma_f32_16x16x4_f32(
        /*neg_a=*/false, a, /*neg_b=*/false, bm,
        /*c_mod=*/(short)0, acc, /*reuse_a=*/false, /*reuse_b=*/false);
  }

  const int colN = lane & 15;
  const int rbase = (tile << 4) + ((lane >> 4) << 3);
#pragma unroll
  for (int r = 0; r < 8; ++r) {
    logits[(size_t)(rbase + r) * 16 + colN] = acc[r];
  }
}

// ---------------------------------------------------------------------------
// Kernel 4: one thread per image: +bias, log_softmax over 10 classes.
// ---------------------------------------------------------------------------
__global__ void __launch_bounds__(256)
lsm_kernel(const float* __restrict__ logits, const float* __restrict__ bias,
           float* __restrict__ out, int Bn) {
  const int b = blockIdx.x * blockDim.x + threadIdx.x;
  if (b >= Bn) return;
  float z[10];
  float m = -INFINITY;
#pragma unroll
  for (int n = 0; n < 10; ++n) {
    z[n] = logits[(size_t)b * 16 + n] + bias[n];
    m = fmaxf(m, z[n]);
  }
  float se = 0.f;
#pragma unroll
  for (int n = 0; n < 10; ++n) se += __expf(z[n] - m);
  const float lse = m + __logf(se);
#pragma unroll
  for (int n = 0; n < 10; ++n) out[(size_t)b * 10 + n] = z[n] - lse;
}

// ---------------------------------------------------------------------------
extern "C" void kernel_launch(void* const* d_in, const int* in_sizes, int n_in,
                              void* d_out, int out_size, void* d_ws, size_t ws_size,
                              hipStream_t stream) {
  const float* x  = (const float*)d_in[0];   // (B, 784)
  const float* rl = (const float*)d_in[1];   // (5,)
  const float* W  = (const float*)d_in[2];   // (10, 784)
  const float* bv = (const float*)d_in[3];   // (10,)
  float* out = (float*)d_out;                // (B, 10)

  const int B = in_sizes[0] / 784;           // 4096
  const int P = B * 196;                     // total patches

  // Workspace carve-up (floats):
  //   feats/weighted : P*4     (scaled in place by attn_kernel)
  //   raw            : P
  //   logits (N=16)  : B*16
  float* feats  = (float*)d_ws;
  float* raw    = feats + (size_t)P * 4;
  float* logits = raw + (size_t)P;

  // 1) quantum circuits: feats + attention raws
  {
    const int threads = 256;
    const int blocks = (P + threads - 1) / threads;
    quanv_kernel<<<blocks, threads, 0, stream>>>(x, rl, feats, raw, P);
  }
  // 2) per-image softmax over patches; feats -> weighted in place
  attn_kernel<<<B, 256, 0, stream>>>(raw, feats);
  // 3) WMMA f32 GEMM: logits = weighted @ W^T (padded to N=16)
  {
    const int tiles = B / 16;                // B is a multiple of 16
    gemm_wmma_kernel<<<tiles / 4, 128, 0, stream>>>(feats, W, logits);
  }
  // 4) bias + log_softmax
  lsm_kernel<<<(B + 255) / 256, 256, 0, stream>>>(logits, bv, out, B);
}